// SpatialTransformer3D_78254304133669
// MI455X (gfx1250) — compile-verified
//
#include <hip/hip_runtime.h>

#define AS1 __attribute__((address_space(1)))
#define AS3 __attribute__((address_space(3)))

#ifndef __has_builtin
#define __has_builtin(x) 0
#endif

// Async global->LDS path is a gfx1250 device-side feature; only probe it in
// the device compilation pass (host pass has no amdgcn builtins).
#if defined(__HIP_DEVICE_COMPILE__) && __has_builtin(__builtin_amdgcn_global_load_async_to_lds_b32)
#define USE_ASYNC 1
#else
#define USE_ASYNC 0
#endif

namespace {
constexpr int Bn  = 2;
constexpr int Dd  = 160;
constexpr int Hh  = 160;
constexpr int Ww  = 160;
constexpr int HW  = Hh * Ww;            // 25600
constexpr int DHW = Dd * HW;            // 4,096,000
constexpr int THREADS = 256;            // 8 wave32 per block
constexpr int ITER    = 4;              // voxels per thread
constexpr int CHUNK   = THREADS * ITER; // 1024; divides DHW exactly -> no batch straddle
}

// One voxel: trilinear gather from the (L2-resident) template.
__device__ __forceinline__ void process_one(const float* __restrict__ tb,
                                            float* __restrict__ ob,
                                            int s, float dx, float dy, float dz)
{
    // s = z*H*W + y*W + x
    int x = s % Ww;
    int t = s / Ww;
    int y = t % Hh;
    int z = t / Hh;

    // Exact simplification of the reference's grid+scale algebra:
    // ix = clip(x + dx, 0, W-1), etc.
    float ix = fminf(fmaxf((float)x + dx, 0.0f), (float)(Ww - 1));
    float iy = fminf(fmaxf((float)y + dy, 0.0f), (float)(Hh - 1));
    float iz = fminf(fmaxf((float)z + dz, 0.0f), (float)(Dd - 1));

    float x0f = floorf(ix), y0f = floorf(iy), z0f = floorf(iz);
    float wx = ix - x0f, wy = iy - y0f, wz = iz - z0f;

    int x0 = (int)x0f, y0 = (int)y0f, z0 = (int)z0f;
    int x1 = min(x0 + 1, Ww - 1);
    int y1 = min(y0 + 1, Hh - 1);
    int z1 = min(z0 + 1, Dd - 1);

    int a00 = z0 * HW + y0 * Ww;
    int a01 = z0 * HW + y1 * Ww;
    int a10 = z1 * HW + y0 * Ww;
    int a11 = z1 * HW + y1 * Ww;

    // 8 gathers; template is ~33MB total -> lives in the 192MB L2.
    float v000 = tb[a00 + x0], v001 = tb[a00 + x1];
    float v010 = tb[a01 + x0], v011 = tb[a01 + x1];
    float v100 = tb[a10 + x0], v101 = tb[a10 + x1];
    float v110 = tb[a11 + x0], v111 = tb[a11 + x1];

    float c00 = fmaf(wx, v001 - v000, v000);
    float c01 = fmaf(wx, v011 - v010, v010);
    float c10 = fmaf(wx, v101 - v100, v100);
    float c11 = fmaf(wx, v111 - v110, v110);
    float c0  = fmaf(wy, c01 - c00, c00);
    float c1  = fmaf(wy, c11 - c10, c10);
    float r   = fmaf(wz, c1 - c0, c0);

    // Streamed output: non-temporal so it doesn't evict the template from L2.
    __builtin_nontemporal_store(r, ob + s);
}

__global__ void __launch_bounds__(THREADS)
st3d_warp_kernel(const float* __restrict__ tpl,
                 const float* __restrict__ disp,
                 float* __restrict__ out)
{
    const int tid   = threadIdx.x;
    const int base  = blockIdx.x * CHUNK;       // < 8,192,000 : fits in int
    const int b     = base / DHW;               // block-uniform (SALU)
    const int sbase = base - b * DHW;

    const float* __restrict__ d0 = disp + (size_t)b * 3 * DHW; // [3][DHW] for this batch
    const float* __restrict__ tb = tpl  + (size_t)b * DHW;
    float*       __restrict__ ob = out  + (size_t)b * DHW;

#if USE_ASYNC
    // Double-buffered displacement staging via CDNA5 async global->LDS
    // (GLOBAL_LOAD_ASYNC_TO_LDS_B32, tracked with ASYNCcnt). Each lane writes
    // and later reads only its own LDS slot -> no barrier, only s_wait_asynccnt.
    __shared__ float sd[2][3][THREADS];

    {   // prologue: stage iteration 0 into buffer 0
        int off = sbase + tid;
        __builtin_amdgcn_global_load_async_to_lds_b32((AS1 int*)(d0 + off),
                                                      (AS3 int*)&sd[0][0][tid], 0, 0);
        __builtin_amdgcn_global_load_async_to_lds_b32((AS1 int*)(d0 + DHW + off),
                                                      (AS3 int*)&sd[0][1][tid], 0, 0);
        __builtin_amdgcn_global_load_async_to_lds_b32((AS1 int*)(d0 + 2 * DHW + off),
                                                      (AS3 int*)&sd[0][2][tid], 0, 0);
    }

#pragma unroll
    for (int i = 0; i < ITER - 1; ++i) {
        // Prefetch next iteration into the other buffer (3 more async ops in flight).
        const int nbuf = (i + 1) & 1;
        const int noff = sbase + (i + 1) * THREADS + tid;
        __builtin_amdgcn_global_load_async_to_lds_b32((AS1 int*)(d0 + noff),
                                                      (AS3 int*)&sd[nbuf][0][tid], 0, 0);
        __builtin_amdgcn_global_load_async_to_lds_b32((AS1 int*)(d0 + DHW + noff),
                                                      (AS3 int*)&sd[nbuf][1][tid], 0, 0);
        __builtin_amdgcn_global_load_async_to_lds_b32((AS1 int*)(d0 + 2 * DHW + noff),
                                                      (AS3 int*)&sd[nbuf][2][tid], 0, 0);

        // 6 outstanding; <=3 means the current iteration's 3 loads have landed.
        asm volatile("s_wait_asynccnt 3" ::: "memory");

        const int cbuf = i & 1;
        process_one(tb, ob, sbase + i * THREADS + tid,
                    sd[cbuf][0][tid], sd[cbuf][1][tid], sd[cbuf][2][tid]);
    }

    asm volatile("s_wait_asynccnt 0" ::: "memory");
    {
        const int cbuf = (ITER - 1) & 1;
        process_one(tb, ob, sbase + (ITER - 1) * THREADS + tid,
                    sd[cbuf][0][tid], sd[cbuf][1][tid], sd[cbuf][2][tid]);
    }
#else
    // Fallback: plain coalesced non-temporal loads (still NT so the streamed
    // displacement field doesn't pollute L2).
#pragma unroll
    for (int i = 0; i < ITER; ++i) {
        int off = sbase + i * THREADS + tid;
        float dx = __builtin_nontemporal_load(d0 + off);
        float dy = __builtin_nontemporal_load(d0 + DHW + off);
        float dz = __builtin_nontemporal_load(d0 + 2 * DHW + off);
        process_one(tb, ob, off, dx, dy, dz);
    }
#endif
}

extern "C" void kernel_launch(void* const* d_in, const int* in_sizes, int n_in,
                              void* d_out, int out_size, void* d_ws, size_t ws_size,
                              hipStream_t stream) {
    (void)in_sizes; (void)n_in; (void)out_size; (void)d_ws; (void)ws_size;
    const float* tpl  = (const float*)d_in[0]; // (B,1,D,H,W) f32
    const float* disp = (const float*)d_in[1]; // (B,3,D,H,W) f32
    float* out = (float*)d_out;                // (B,1,D,H,W) f32

    const int total  = Bn * DHW;               // 8,192,000
    const int blocks = total / CHUNK;          // 8000 (exact)
    st3d_warp_kernel<<<blocks, THREADS, 0, stream>>>(tpl, disp, out);
}